// eppConcluer_26139170964130
// MI455X (gfx1250) — compile-verified
//
#include <hip/hip_runtime.h>
#include <math.h>

typedef float v2f __attribute__((ext_vector_type(2)));
typedef float v8f __attribute__((ext_vector_type(8)));

// ---------------- workspace layout (floats) ----------------
#define WS_KINV 0        // 9
#define WS_M4   16       // 48*48 = 2304  -> [16,2320)   (row 45 = sum_q)
#define WS_F    2320     // 9
#define WS_ANG  2329     // 3
#define WS_T    2332     // 3
#define WS_LOSS 2335     // 1
#define WS_END  2336

#define ITNUM_ 100
#define LAP_   0.01f
#define LR_    0.1f
#define SQRT2_ 1.41421356237309515f

// unrank s in [0,45) -> pair (a<=b), a,b in [0,9)
__device__ __forceinline__ void unrank45(int s, int* A, int* B) {
  int a = 0, rem = s;
  while (rem >= 9 - a) { rem -= 9 - a; ++a; }
  *A = a; *B = a + rem;
}

__device__ __forceinline__ float sel3(int i, float a, float b, float c) {
  return (i == 0) ? a : ((i == 1) ? b : c);
}

__device__ __forceinline__ void m3mul(const float* A, const float* B, float* C) {
#pragma unroll
  for (int i = 0; i < 3; ++i)
#pragma unroll
    for (int j = 0; j < 3; ++j)
      C[i*3+j] = A[i*3+0]*B[0*3+j] + A[i*3+1]*B[1*3+j] + A[i*3+2]*B[2*3+j];
}

__device__ __forceinline__ float dot9(const float* a, const float* b) {
  float s = 0.f;
#pragma unroll
  for (int i = 0; i < 9; ++i) s += a[i]*b[i];
  return s;
}

__device__ void build_rot(float a0, float a1, float a2,
                          float* R, float* Rxd, float* Ryd, float* Rzd) {
  float cx = cosf(a0), sx = sinf(a0);
  float cy = cosf(a1), sy = sinf(a1);
  float cz = cosf(a2), sz = sinf(a2);
  float rotx[9]  = {1,0,0, 0,cx,-sx, 0,sx,cx};
  float roty[9]  = {cy,0,sy, 0,1,0, -sy,0,cy};
  float rotz[9]  = {cz,-sz,0, sz,cz,0, 0,0,1};
  float rotxd[9] = {0,0,0, 0,-sx,-cx, 0,cx,-sx};
  float rotyd[9] = {-sy,0,cy, 0,0,0, -cy,0,-sy};
  float rotzd[9] = {-sz,-cz,0, cz,-sz,0, 0,0,0};
  float zy[9], tmp[9];
  m3mul(rotz, roty, zy);
  m3mul(zy, rotx, R);
  m3mul(zy, rotxd, Rxd);
  m3mul(rotz, rotyd, tmp); m3mul(tmp, rotx, Ryd);
  m3mul(rotzd, roty, tmp); m3mul(tmp, rotx, Rzd);
}

// ---------------- kernel 0: init ----------------
__global__ void epp_init_kernel(float* __restrict__ ws, const float* __restrict__ K) {
  int tid = threadIdx.x;
  for (int i = WS_M4 + tid; i < WS_END; i += blockDim.x) ws[i] = 0.0f;
  if (tid == 0) {
    float det = K[0]*(K[4]*K[8]-K[5]*K[7]) - K[1]*(K[3]*K[8]-K[5]*K[6]) + K[2]*(K[3]*K[7]-K[4]*K[6]);
    float id = 1.0f/det;
    ws[WS_KINV+0] = (K[4]*K[8]-K[5]*K[7])*id;
    ws[WS_KINV+1] = (K[2]*K[7]-K[1]*K[8])*id;
    ws[WS_KINV+2] = (K[1]*K[5]-K[2]*K[4])*id;
    ws[WS_KINV+3] = (K[5]*K[6]-K[3]*K[8])*id;
    ws[WS_KINV+4] = (K[0]*K[8]-K[2]*K[6])*id;
    ws[WS_KINV+5] = (K[2]*K[3]-K[0]*K[5])*id;
    ws[WS_KINV+6] = (K[3]*K[7]-K[4]*K[6])*id;
    ws[WS_KINV+7] = (K[1]*K[6]-K[0]*K[7])*id;
    ws[WS_KINV+8] = (K[0]*K[4]-K[1]*K[3])*id;
  }
}

// ---------------- kernel 1: 2nd+4th moments via symmetric WMMA SYRK ----------------
// qhat in R^48: comps 0..44 = w_s*m_a*m_b, comp 45 = 1 (valid point), 46/47 = 0.
// M4(48x48) += sum_n qhat qhat^T, upper-triangle tiles only (mirrored on writeback).
// Row 45 of M4 = sum_n qhat (2nd moment), M4[45][45] = N.
__global__ void epp_moment4_kernel(const float* __restrict__ p1raw,
                                   const float* __restrict__ p2raw,
                                   float* __restrict__ ws, int N) {
  const int lane = threadIdx.x & 31;
  const int wid  = blockIdx.x * (blockDim.x >> 5) + (threadIdx.x >> 5);
  const int W    = gridDim.x * (blockDim.x >> 5);
  const int l15  = lane & 15;
  const bool hi  = lane >= 16;
  const int shift = hi ? 1 : 0;

  float Ki[9];
#pragma unroll
  for (int i = 0; i < 9; ++i) Ki[i] = ws[WS_KINV + i];

  // per-lane component descriptors for components {l15, 16+l15, 32+l15}
  int ua[3], va[3], wa[3], xa[3];
  float wt[3];
  bool one45[3];
#pragma unroll
  for (int ci = 0; ci < 3; ++ci) {
    int c = ci*16 + l15;
    one45[ci] = (c == 45);
    if (c < 45) {
      int a, b; unrank45(c, &a, &b);
      ua[ci] = a/3; va[ci] = a%3; wa[ci] = b/3; xa[ci] = b%3;
      wt[ci] = (a == b) ? 1.0f : SQRT2_;
    } else { ua[ci]=va[ci]=wa[ci]=xa[ci]=0; wt[ci] = 0.0f; }
  }

  v8f acc00 = (v8f){0.f,0.f,0.f,0.f,0.f,0.f,0.f,0.f};
  v8f acc01 = acc00, acc02 = acc00, acc11 = acc00, acc12 = acc00, acc22 = acc00;

  const int ngroups = (N + 3) >> 2;
#pragma unroll 2
  for (int g = wid; g < ngroups; g += W) {
    int base = g * 4;
    int myn  = base + (lane & 3);
    int nc   = (myn < N) ? myn : (N - 1);
    float X1 = p1raw[nc], Y1 = p1raw[N + nc];
    float X2 = p2raw[nc], Y2 = p2raw[N + nc];
    // normalized coords (Kinv applied) for this lane's own point
    float a1x = Ki[0]*X1 + Ki[1]*Y1 + Ki[2];
    float a1y = Ki[3]*X1 + Ki[4]*Y1 + Ki[5];
    float a1z = Ki[6]*X1 + Ki[7]*Y1 + Ki[8];
    float a2x = Ki[0]*X2 + Ki[1]*Y2 + Ki[2];
    float a2y = Ki[3]*X2 + Ki[4]*Y2 + Ki[5];
    float a2z = Ki[6]*X2 + Ki[7]*Y2 + Ki[8];

    // Each lane only needs points {0,1,2}+shift of this 4-group.
    float qp[3][3];
#pragma unroll
    for (int j = 0; j < 3; ++j) {
      int p = j + shift;
      float p1x = __shfl(a1x, p), p1y = __shfl(a1y, p), p1z = __shfl(a1z, p);
      float p2x = __shfl(a2x, p), p2y = __shfl(a2y, p), p2z = __shfl(a2z, p);
      float vk = (base + p < N) ? 1.0f : 0.0f;
#pragma unroll
      for (int ci = 0; ci < 3; ++ci) {
        float s2u = sel3(ua[ci], p2x, p2y, p2z);
        float s1v = sel3(va[ci], p1x, p1y, p1z);
        float s2w = sel3(wa[ci], p2x, p2y, p2z);
        float s1x = sel3(xa[ci], p1x, p1y, p1z);
        float val = (wt[ci]*vk) * (s2u*s1v) * (s2w*s1x);
        qp[j][ci] = one45[ci] ? vk : val;
      }
    }

    v2f A[3], B[3];
#pragma unroll
    for (int ci = 0; ci < 3; ++ci) {
      // A 16x4 f32: lanes0-15 = K{0,1}; lanes16-31 = K{2,3}
      A[ci].x = hi ? qp[1][ci] : qp[0][ci];
      A[ci].y = hi ? qp[2][ci] : qp[1][ci];
      // B 4x16 f32: VGPR0 = rows K{0|1}, VGPR1 = rows K{2|3} by half
      B[ci].x = qp[0][ci];
      B[ci].y = qp[2][ci];
    }

    acc00 = __builtin_amdgcn_wmma_f32_16x16x4_f32(false, A[0], false, B[0], (short)0, acc00, false, false);
    acc01 = __builtin_amdgcn_wmma_f32_16x16x4_f32(false, A[0], false, B[1], (short)0, acc01, false, false);
    acc02 = __builtin_amdgcn_wmma_f32_16x16x4_f32(false, A[0], false, B[2], (short)0, acc02, false, false);
    acc11 = __builtin_amdgcn_wmma_f32_16x16x4_f32(false, A[1], false, B[1], (short)0, acc11, false, false);
    acc12 = __builtin_amdgcn_wmma_f32_16x16x4_f32(false, A[1], false, B[2], (short)0, acc12, false, false);
    acc22 = __builtin_amdgcn_wmma_f32_16x16x4_f32(false, A[2], false, B[2], (short)0, acc22, false, false);
  }

  // writeback: C layout: lane L, VGPR r -> row = r + 8*(L>=16), col = L&15.
  // Mirror off-diagonal tiles (SYRK symmetry).
  float* M4 = ws + WS_M4;
  const int rofs = hi ? 8 : 0;
  {
    const v8f* accs[6] = {&acc00, &acc01, &acc02, &acc11, &acc12, &acc22};
    const int tis[6] = {0,0,0,1,1,2};
    const int tjs[6] = {0,1,2,1,2,2};
#pragma unroll
    for (int tt = 0; tt < 6; ++tt) {
      int ti = tis[tt], tj = tjs[tt];
#pragma unroll
      for (int r = 0; r < 8; ++r) {
        int grow = ti*16 + r + rofs;
        int gcol = tj*16 + l15;
        float v = (*accs[tt])[r];
        atomicAdd(&M4[grow*48 + gcol], v);
        if (ti != tj) atomicAdd(&M4[gcol*48 + grow], v);
      }
    }
  }
}

// ---------------- kernel 2: 100 Gauss-Newton iterations (single block) ----------------
__global__ void epp_iterate_kernel(float* __restrict__ ws, int N) {
  __shared__ float sQ2[81], sP[81], sU[9], sV[54], sUh[45], sT4[45], sY[72];
  __shared__ float sStop[1];
  const int tid = threadIdx.x;
  const float* M4 = ws + WS_M4;

  if (tid < 45) {                       // unpack 2nd moment once (row 45 of M4)
    int a, b; unrank45(tid, &a, &b);
    float w = (a == b) ? 1.0f : SQRT2_;
    float v = M4[45*48 + tid] / w;
    sQ2[a*9+b] = v; sQ2[b*9+a] = v;
  }
  __syncthreads();

  float ang0 = 0.f, ang1 = 0.f, ang2 = 0.f;
  float t0 = 0.f, t1 = 0.f, t2 = -0.95f;
  float minloss = 1e10f;
  const float Nf = (float)N, invN = 1.0f/Nf, invN2 = invN*invN;

  for (int it = 0; it < ITNUM_; ++it) {
    if (tid == 0) {
      float R[9], Rxd[9], Ryd[9], Rzd[9];
      build_rot(ang0, ang1, ang2, R, Rxd, Ryd, Rzd);
      float T[9] = {0,-t2,t1, t2,0,-t0, -t1,t0,0};
      float U[9], V0[9], V1[9], V2[9];
      m3mul(T, R, U);
      m3mul(T, Rxd, V0); m3mul(T, Ryd, V1); m3mul(T, Rzd, V2);
      float V3[9] = {0,0,0, -R[6],-R[7],-R[8], R[3],R[4],R[5]};     // DT0@R
      float V4[9] = {R[6],R[7],R[8], 0,0,0, -R[0],-R[1],-R[2]};     // DT1@R
      float V5[9] = {-R[3],-R[4],-R[5], R[0],R[1],R[2], 0,0,0};     // DT2@R
#pragma unroll
      for (int i = 0; i < 9; ++i) {
        sU[i] = U[i];
        sV[0*9+i]=V0[i]; sV[1*9+i]=V1[i]; sV[2*9+i]=V2[i];
        sV[3*9+i]=V3[i]; sV[4*9+i]=V4[i]; sV[5*9+i]=V5[i];
      }
    }
    __syncthreads();
    if (tid < 45) {                     // uhat
      int a, b; unrank45(tid, &a, &b);
      float w = (a == b) ? 1.0f : SQRT2_;
      sUh[tid] = w * sU[a] * sU[b];
    }
    __syncthreads();
    if (tid < 45) {                     // t4 = M4 * uhat
      const float* row = M4 + tid*48;
      float acc = 0.f;
      for (int s = 0; s < 45; ++s) acc += row[s]*sUh[s];
      sT4[tid] = acc;
    }
    __syncthreads();
    if (tid < 45) {                     // unpack P[a][b] = sum r^2 m_a m_b
      int a, b; unrank45(tid, &a, &b);
      float w = (a == b) ? 1.0f : SQRT2_;
      float v = sT4[tid] / w;
      sP[a*9+b] = v; sP[b*9+a] = v;
    }
    __syncthreads();
    if (tid < 72) {                     // y_k = P v_k, y6 = P u, y7 = Q2 u
      int vi = tid/9, rowi = tid%9;
      const float* vec = (vi < 6) ? (sV + vi*9) : sU;
      const float* Mx  = (vi == 7) ? sQ2 : sP;
      float acc = 0.f;
#pragma unroll
      for (int c = 0; c < 9; ++c) acc += Mx[rowi*9+c]*vec[c];
      sY[vi*9+rowi] = acc;
    }
    __syncthreads();
    if (tid == 0) {
      float S2[36], S3[6], S1[6];
      for (int k = 0; k < 6; ++k) {
        S3[k] = dot9(sU, sY + k*9);
        S1[k] = dot9(sV + k*9, sY + 63);
        for (int l = 0; l < 6; ++l) S2[k*6+l] = dot9(sV + k*9, sY + l*9);
      }
      float Sr2 = dot9(sU, sY + 63);
      float tn = sqrtf(t0*t0 + t1*t1 + t2*t2);
      float rb = tn - 1.0f;
      float bs = 2.0f*LAP_*rb/tn;
      float bb[6] = {0.f,0.f,0.f, bs*t0, bs*t1, bs*t2};
      float cc = LAP_*rb*rb;
      float M6[36], g6[6];
      for (int k = 0; k < 6; ++k) {
        g6[k] = (2.0f*S3[k] + 2.0f*cc*S1[k] + Sr2*bb[k] + Nf*cc*bb[k]) * invN2;
        for (int l = 0; l < 6; ++l)
          M6[k*6+l] = (4.0f*S2[k*6+l] + 2.0f*(S1[k]*bb[l] + bb[k]*S1[l]) + Nf*bb[k]*bb[l]) * invN2;
      }
      float curloss = Sr2*invN + cc;
      // 6x6 Gauss-Jordan inverse with partial pivoting
      float A_[36], Inv[36];
      for (int i = 0; i < 36; ++i) { A_[i] = M6[i]; Inv[i] = 0.f; }
      for (int i = 0; i < 6; ++i) Inv[i*6+i] = 1.f;
      for (int col = 0; col < 6; ++col) {
        int piv = col; float pv = fabsf(A_[col*6+col]);
        for (int rr = col+1; rr < 6; ++rr) { float v = fabsf(A_[rr*6+col]); if (v > pv) { pv = v; piv = rr; } }
        if (piv != col)
          for (int j = 0; j < 6; ++j) {
            float tA = A_[col*6+j]; A_[col*6+j] = A_[piv*6+j]; A_[piv*6+j] = tA;
            float tI = Inv[col*6+j]; Inv[col*6+j] = Inv[piv*6+j]; Inv[piv*6+j] = tI;
          }
        float rd = 1.0f / A_[col*6+col];
        for (int j = 0; j < 6; ++j) { A_[col*6+j] *= rd; Inv[col*6+j] *= rd; }
        for (int rr = 0; rr < 6; ++rr) if (rr != col) {
          float f = A_[rr*6+col];
          for (int j = 0; j < 6; ++j) { A_[rr*6+j] -= f*A_[col*6+j]; Inv[rr*6+j] -= f*Inv[col*6+j]; }
        }
      }
      float maxerr = 0.f;
      for (int i = 0; i < 6; ++i)
        for (int j = 0; j < 6; ++j) {
          float e = 0.f;
          for (int k = 0; k < 6; ++k) e += Inv[i*6+k]*M6[k*6+j];
          e -= (i == j) ? 1.f : 0.f;
          maxerr = fmaxf(maxerr, fabsf(e));
        }
      bool bad  = maxerr > 0.1f;
      bool stop = bad || (curloss > minloss);
      if (stop) {
        sStop[0] = 1.0f;
      } else {
        float up[6];
        for (int k = 0; k < 6; ++k) {
          float s = 0.f;
          for (int l = 0; l < 6; ++l) s += Inv[k*6+l]*g6[l];
          up[k] = s;
        }
        ang0 -= LR_*up[0]; ang1 -= LR_*up[1]; ang2 -= LR_*up[2];
        t0   -= LR_*up[3]; t1   -= LR_*up[4]; t2   -= LR_*up[5];
        minloss = curloss;
        sStop[0] = 0.0f;
      }
    }
    __syncthreads();
    if (sStop[0] != 0.0f) break;
    __syncthreads();
  }

  if (tid == 0) {
    float R[9], Rxd[9], Ryd[9], Rzd[9];
    build_rot(ang0, ang1, ang2, R, Rxd, Ryd, Rzd);
    float tn = sqrtf(t0*t0 + t1*t1 + t2*t2);
    float u0 = t0/tn, u1 = t1/tn, u2 = t2/tn;
    float Tn[9] = {0,-u2,u1, u2,0,-u0, -u1,u0,0};
    float E[9]; m3mul(Tn, R, E);
    float Ki[9], KT[9];
    for (int i = 0; i < 9; ++i) Ki[i] = ws[WS_KINV + i];
    for (int i = 0; i < 3; ++i) for (int j = 0; j < 3; ++j) KT[i*3+j] = Ki[j*3+i];
    float tmp[9], F[9];
    m3mul(KT, E, tmp); m3mul(tmp, Ki, F);
    for (int i = 0; i < 9; ++i) ws[WS_F + i] = F[i];
    ws[WS_ANG+0] = ang0; ws[WS_ANG+1] = ang1; ws[WS_ANG+2] = ang2;
    ws[WS_T+0]   = t0;   ws[WS_T+1]   = t1;   ws[WS_T+2]   = t2;
  }
}

// ---------------- kernel 3: final loss = mean |p2^T F p1| ----------------
__global__ void epp_loss_kernel(const float* __restrict__ p1raw,
                                const float* __restrict__ p2raw,
                                float* __restrict__ ws, int N) {
  float F[9];
#pragma unroll
  for (int i = 0; i < 9; ++i) F[i] = ws[WS_F + i];
  float acc = 0.f;
  for (int n = blockIdx.x*blockDim.x + threadIdx.x; n < N; n += gridDim.x*blockDim.x) {
    float x1 = p1raw[n], y1 = p1raw[N + n];
    float x2 = p2raw[n], y2 = p2raw[N + n];
    float r0 = F[0]*x1 + F[1]*y1 + F[2];
    float r1 = F[3]*x1 + F[4]*y1 + F[5];
    float r2 = F[6]*x1 + F[7]*y1 + F[8];
    acc += fabsf(x2*r0 + y2*r1 + r2);
  }
  for (int off = 16; off > 0; off >>= 1) acc += __shfl_down(acc, off);
  if ((threadIdx.x & 31) == 0) atomicAdd(&ws[WS_LOSS], acc);
}

// ---------------- kernel 4: writeout ----------------
__global__ void epp_writeout_kernel(const float* __restrict__ ws, float* __restrict__ out, int N) {
  int tid = threadIdx.x;
  if (tid < 9)       out[tid] = ws[WS_F + tid];
  else if (tid < 12) out[tid] = ws[WS_ANG + tid - 9];
  else if (tid < 15) out[tid] = ws[WS_T + tid - 12];
  else if (tid == 15) out[15] = ws[WS_LOSS] / (float)N;
}

extern "C" void kernel_launch(void* const* d_in, const int* in_sizes, int n_in,
                              void* d_out, int out_size, void* d_ws, size_t ws_size,
                              hipStream_t stream) {
  const float* pts1 = (const float*)d_in[0];
  const float* pts2 = (const float*)d_in[1];
  const float* K    = (const float*)d_in[2];
  float* out = (float*)d_out;
  float* ws  = (float*)d_ws;
  const int N = in_sizes[0] / 3;

  epp_init_kernel<<<1, 256, 0, stream>>>(ws, K);
  epp_moment4_kernel<<<128, 256, 0, stream>>>(pts1, pts2, ws, N);
  epp_iterate_kernel<<<1, 128, 0, stream>>>(ws, N);
  epp_loss_kernel<<<256, 256, 0, stream>>>(pts1, pts2, ws, N);
  epp_writeout_kernel<<<1, 32, 0, stream>>>(ws, out, N);
}